// MeanAggregator_56349970923547
// MI455X (gfx1250) — compile-verified
//
#include <hip/hip_runtime.h>
#include <hip/hip_bf16.h>
#include <stdint.h>

#ifndef __has_builtin
#define __has_builtin(x) 0
#endif

#define K_NEIGH 6
#define DIM 256
#define TILE_NODES 16
#define TILE_ROWS (K_NEIGH * TILE_NODES)   // 96 gathered rows per tile
#define CHUNK_ROWS 16                      // rows staged per async chunk (16 KB)
#define NUM_CHUNKS (TILE_ROWS / CHUNK_ROWS) // 6
#define ROWF 260                           // padded LDS row: 256 + 4 floats (bank-conflict pad, keeps 16B align)
#define THREADS 128                        // 4 wave32 waves
#define NWAVES (THREADS / 32)

typedef float v2f __attribute__((ext_vector_type(2)));
typedef float v8f __attribute__((ext_vector_type(8)));
typedef int   v4i __attribute__((vector_size(16)));   // matches builtin param type

#if __has_builtin(__builtin_amdgcn_wmma_f32_16x16x4_f32)
#define HAVE_WMMA_F32 1
#endif
#if __has_builtin(__builtin_amdgcn_global_load_async_to_lds_b128)
#define HAVE_ASYNC_B128 1
#endif
#if __has_builtin(__builtin_amdgcn_s_wait_asynccnt)
#define HAVE_WAIT_ASYNC 1
#endif

// ---- CDNA5 async global->LDS 16-byte copy (ASYNCcnt-tracked) ----
__device__ __forceinline__ void async_copy16(const float* gsrc, float* ldst) {
#if defined(HAVE_ASYNC_B128)
  __builtin_amdgcn_global_load_async_to_lds_b128(
      (__attribute__((address_space(1))) v4i*)(uintptr_t)gsrc,
      (__attribute__((address_space(3))) v4i*)(uint32_t)(uintptr_t)ldst,
      /*offset=*/0, /*cpol=*/0);
#else
  // Fallback: raw CDNA5 encoding. VDST = LDS byte address, VADDR = 64-bit global VA.
  asm volatile("global_load_async_to_lds_b128 %0, %1, off"
               :
               : "v"((uint32_t)(uintptr_t)ldst), "v"(gsrc)
               : "memory");
#endif
}

__device__ __forceinline__ void wait_async_zero() {
#if defined(HAVE_WAIT_ASYNC)
  __builtin_amdgcn_s_wait_asynccnt(0);
#else
  asm volatile("s_wait_asynccnt 0" ::: "memory");
#endif
}

__global__ __launch_bounds__(THREADS) void
mean_agg_wmma_kernel(const float* __restrict__ feats,
                     const int* __restrict__ nidx,
                     float* __restrict__ out, int nB) {
  __shared__ int sIdx[TILE_ROWS];
#if defined(HAVE_WMMA_F32)
  __shared__ float sBuf[2][CHUNK_ROWS * ROWF];   // 2 x 16.25 KB double buffer
#endif

  const int tid   = threadIdx.x;
  const int node0 = blockIdx.x * TILE_NODES;

  // Stage this tile's 96 neighbor indices into LDS.
  if (tid < TILE_ROWS) {
    const int node = node0 + tid / K_NEIGH;
    sIdx[tid] = (node < nB) ? nidx[node * K_NEIGH + (tid % K_NEIGH)] : 0;
  }
  __syncthreads();

  const int   lane = tid & 31;
  const int   wave = tid >> 5;
  const int   hi   = lane >> 4;   // half-wave select (K/M split in WMMA layouts)
  const int   m    = lane & 15;   // M row (A/C) == N col (B) for this lane
  const float W    = 1.0f / (float)K_NEIGH;

#if defined(HAVE_WMMA_F32)
  // Async-stage one 16-row chunk (16 KB): 128 threads x 8 x 16B.
  auto stage = [&](int c, float* buf) {
#pragma unroll
    for (int l = 0; l < 8; ++l) {
      const int q   = l * THREADS + tid;  // 0..1023 : 16 rows x 64 16B-segments
      const int r   = q >> 6;             // row within chunk
      const int seg = q & 63;             // 16B segment within row
      const int gi  = sIdx[c * CHUNK_ROWS + r];
      async_copy16(feats + (size_t)gi * DIM + seg * 4,
                   buf + r * ROWF + seg * 4);
    }
  };

  v8f acc[4] = {};   // 4 col-tiles of 16x16 f32 per wave (32 VGPRs)

  stage(0, sBuf[0]);
  wait_async_zero();
  __syncthreads();

  for (int c = 0; c < NUM_CHUNKS; ++c) {
    if (c + 1 < NUM_CHUNKS) stage(c + 1, sBuf[(c + 1) & 1]);  // overlap copy w/ compute
    const float* buf = sBuf[c & 1];

#pragma unroll
    for (int s = 0; s < CHUNK_ROWS / 4; ++s) {
      // A (16x4 f32): lane<16 holds K=0,1; lane>=16 holds K=2,3; M = lane&15.
      // Weight pattern is static: gathered row R belongs to node R/6 with weight 1/6.
      const int R0 = c * CHUNK_ROWS + 4 * s + 2 * hi;
      v2f a;
      a.x = ((R0 / K_NEIGH) == m) ? W : 0.0f;
      a.y = (((R0 + 1) / K_NEIGH) == m) ? W : 0.0f;

      // B (4x16 f32): lane<16 holds K=0,1 col N=lane; lane>=16 holds K=2,3.
      const float* brow = buf + (4 * s + 2 * hi) * ROWF + m;
#pragma unroll
      for (int u = 0; u < 4; ++u) {
        const int colbase = (wave * 4 + u) * 16;
        v2f b;
        b.x = brow[colbase];
        b.y = brow[colbase + ROWF];
        acc[u] = __builtin_amdgcn_wmma_f32_16x16x4_f32(
            /*neg_a=*/false, a, /*neg_b=*/false, b,
            /*c_mod=*/(short)0, acc[u], /*reuse_a=*/false, /*reuse_b=*/false);
      }
    }
    wait_async_zero();   // chunk c+1 resident before next iteration reads it
    __syncthreads();
  }

  // C/D layout: VGPR v -> row M = v + 8*hi, col N = lane&15.
#pragma unroll
  for (int u = 0; u < 4; ++u) {
    const int colbase = (wave * 4 + u) * 16;
#pragma unroll
    for (int v = 0; v < 8; ++v) {
      const int row = node0 + v + 8 * hi;
      if (row < nB) out[(size_t)row * DIM + colbase + m] = acc[u][v];
    }
  }
#else
  // Exact-f32 VALU fallback (same launch geometry): one node per wave pass,
  // each lane owns 8 contiguous columns.
  for (int nn = wave; nn < TILE_NODES; nn += NWAVES) {
    const int node = node0 + nn;
    float accv[8] = {0.f, 0.f, 0.f, 0.f, 0.f, 0.f, 0.f, 0.f};
#pragma unroll
    for (int k = 0; k < K_NEIGH; ++k) {
      const float* rowp = feats + (size_t)sIdx[nn * K_NEIGH + k] * DIM + lane * 8;
#pragma unroll
      for (int e = 0; e < 8; ++e) accv[e] += rowp[e];
    }
    if (node < nB) {
      float* op = out + (size_t)node * DIM + lane * 8;
#pragma unroll
      for (int e = 0; e < 8; ++e) op[e] = accv[e] * W;
    }
  }
#endif
}

extern "C" void kernel_launch(void* const* d_in, const int* in_sizes, int n_in,
                              void* d_out, int out_size, void* d_ws, size_t ws_size,
                              hipStream_t stream) {
  (void)in_sizes; (void)n_in; (void)d_ws; (void)ws_size;
  const float* feats = (const float*)d_in[0];
  const int*   nidx  = (const int*)d_in[1];
  float*       out   = (float*)d_out;
  const int nB    = out_size / DIM;                         // 50000
  const int tiles = (nB + TILE_NODES - 1) / TILE_NODES;     // 3125
  mean_agg_wmma_kernel<<<dim3(tiles), dim3(THREADS), 0, stream>>>(feats, nidx, out, nB);
}